// NNDModule_42417097016834
// MI455X (gfx1250) — compile-verified
//
#include <hip/hip_runtime.h>

// Chamfer distance via V_WMMA_F32_16X16X4_F32 (fp32 rank-4 matrix op).
//
// d[i][j] = |p_i|^2 + |q_j|^2 - 2 p_i.q_j  is computed as a single WMMA:
//   A (16x4):  row m   = (-2x1, -2y1, -2z1, 1.0)
//   B (4x16):  col n   = ( x2,   y2,   z2,  |q_n|^2 )
//   C (16x16): C[m][n] = |p_m|^2              (loop invariant per wave)
//   D = A*B + C  ==  16x16 tile of squared distances, full fp32.
//
// Each wave owns one 16-row tile of the "query" set and sweeps all column
// tiles, keeping an elementwise running min of D. Row-mins finish with a
// shfl_xor min-tree over the 16 lanes of each half (D layout: VGPR v, half h
// -> row m = v + 8h, column n = lane & 15). dist2 = same kernel, args swapped.

typedef __attribute__((ext_vector_type(2))) float v2f;
typedef __attribute__((ext_vector_type(8))) float v8f;

__global__ __launch_bounds__(256) void chamfer_rowmin_wmma(
    const float* __restrict__ P,   // [B, Nrows, 3] query points (rows of d)
    const float* __restrict__ Q,   // [B, Mcols, 3] reference points (cols of d)
    float* __restrict__ out,       // [B, Nrows]  out[i] = min_j d(i,j)
    int Nrows, int Mcols, int total_waves)
{
    const int waves_per_block = blockDim.x >> 5;
    const int wave_in_block   = threadIdx.x >> 5;
    const int lane            = threadIdx.x & 31;
    const int wave_id         = blockIdx.x * waves_per_block + wave_in_block;
    if (wave_id >= total_waves) return;   // whole-wave guard: EXEC stays all-1s

    const int tiles_per_batch = Nrows >> 4;
    const int batch    = wave_id / tiles_per_batch;
    const int tile     = wave_id % tiles_per_batch;
    const int row_base = tile << 4;

    const float* p = P + (size_t)batch * Nrows * 3;
    const float* q = Q + (size_t)batch * Mcols * 3;

    const int hi  = lane >> 4;    // half: 0 -> lanes 0-15, 1 -> lanes 16-31
    const int sub = lane & 15;

    // ---- A operand (16x4 fp32, 2 VGPRs):
    //      lanes 0-15:  M=sub, VGPR0=K0=-2x, VGPR1=K1=-2y
    //      lanes 16-31: M=sub, VGPR0=K2=-2z, VGPR1=K3= 1.0
    const int arow = row_base + sub;
    const float ax = p[arow * 3 + 0];
    const float ay = p[arow * 3 + 1];
    const float az = p[arow * 3 + 2];
    v2f A;
    A.x = hi ? (-2.0f * az) : (-2.0f * ax);
    A.y = hi ? 1.0f         : (-2.0f * ay);

    // ---- C operand: C[m][n] = |p_m|^2 ; this lane holds rows m = v + 8*hi.
    v8f C;
#pragma unroll
    for (int v = 0; v < 8; ++v) {
        const int r = row_base + v + 8 * hi;
        const float x = p[r * 3 + 0];
        const float y = p[r * 3 + 1];
        const float z = p[r * 3 + 2];
        C[v] = x * x + y * y + z * z;
    }

    v8f acc;
#pragma unroll
    for (int v = 0; v < 8; ++v) acc[v] = 3.0e38f;

    for (int j0 = 0; j0 < Mcols; j0 += 16) {
        const int col = j0 + sub;
        const float bx = q[col * 3 + 0];
        const float by = q[col * 3 + 1];
        const float bz = q[col * 3 + 2];
        const float sq = bx * bx + by * by + bz * bz;
        // ---- B operand (4x16 fp32, 2 VGPRs), mirrored striping of A:
        //      lanes 0-15:  N=sub, VGPR0=K0=x2, VGPR1=K1=y2
        //      lanes 16-31: N=sub, VGPR0=K2=z2, VGPR1=K3=|q|^2
        v2f Bm;
        Bm.x = hi ? bz : bx;
        Bm.y = hi ? sq : by;

        const v8f D = __builtin_amdgcn_wmma_f32_16x16x4_f32(
            /*neg_a=*/false, A, /*neg_b=*/false, Bm,
            /*c_mod=*/(short)0, C, /*reuse_a=*/false, /*reuse_b=*/false);

#pragma unroll
        for (int v = 0; v < 8; ++v) acc[v] = fminf(acc[v], D[v]);
    }

    // ---- min over the 16 columns held across lanes of this half.
#pragma unroll
    for (int v = 0; v < 8; ++v) {
        float m = acc[v];
        m = fminf(m, __shfl_xor(m, 1, 32));
        m = fminf(m, __shfl_xor(m, 2, 32));
        m = fminf(m, __shfl_xor(m, 4, 32));
        m = fminf(m, __shfl_xor(m, 8, 32));   // stays inside each 16-lane half
        if (sub == v)
            out[(size_t)batch * Nrows + row_base + v + 8 * hi] = m;
    }
}

extern "C" void kernel_launch(void* const* d_in, const int* in_sizes, int n_in,
                              void* d_out, int out_size, void* d_ws, size_t ws_size,
                              hipStream_t stream) {
    (void)n_in; (void)out_size; (void)d_ws; (void)ws_size;
    const float* xyz1 = (const float*)d_in[0];   // [B, N, 3]
    const float* xyz2 = (const float*)d_in[1];   // [B, M, 3]

    const int N = 4096, M = 4096;
    const int B = in_sizes[0] / (N * 3);

    float* dist1 = (float*)d_out;                // [B, N]
    float* dist2 = dist1 + (size_t)B * N;        // [B, M]

    const int threads = 256;
    const int wpb = threads / 32;

    const int waves1 = B * (N / 16);
    chamfer_rowmin_wmma<<<(waves1 + wpb - 1) / wpb, threads, 0, stream>>>(
        xyz1, xyz2, dist1, N, M, waves1);

    const int waves2 = B * (M / 16);
    chamfer_rowmin_wmma<<<(waves2 + wpb - 1) / wpb, threads, 0, stream>>>(
        xyz2, xyz1, dist2, M, N, waves2);
}